// SpatialTransformer_73547019977137
// MI455X (gfx1250) — compile-verified
//
#include <hip/hip_runtime.h>
#include <cstdint>

// Problem constants (from reference setup_inputs)
#define B_  2
#define C_  2
#define D_  160
#define H_  192
#define W_  192

constexpr int HW   = H_ * W_;          // 36864
constexpr int DHW  = D_ * HW;          // 5,898,240
constexpr int NPTS = B_ * DHW;         // 11,796,480 spatial points
constexpr int TILE = 256;              // 8 wave32s / block

// 3D trilinear warp, align_corners=True, border padding.
// Displacement is streamed one tile ahead through LDS via the CDNA5
// async-to-LDS path (ASYNCcnt), double-buffered, no barriers needed
// (each wave stages + consumes only its own lanes' slots).
__global__ __launch_bounds__(TILE) void warp3d_async_kernel(
    const float* __restrict__ src,
    const float* __restrict__ disp,
    float* __restrict__ out,
    int numTiles)
{
    __shared__ float sdisp[2][3][TILE];   // [stage][component z,y,x][lane]

    const int tid = threadIdx.x;
    // Low 32 bits of a flat shared pointer = workgroup-relative LDS byte
    // offset, which is exactly what global_load_async_to_lds_b32's VDST wants.
    const uint32_t ldsBase = (uint32_t)(uintptr_t)&sdisp[0][0][0];

    // Issue 3 async global->LDS loads (dz, dy, dx) for one tile into a stage.
    // OOB lanes are clamped to a valid address (never EXEC-masked) so the
    // wave always issues exactly 3 ASYNCcnt ops -> simple counter discipline.
    auto issue_tile = [&](int tile, int stg) {
        int i = tile * TILE + tid;
        i = (i < NPTS) ? i : (NPTS - 1);
        uint32_t b = (uint32_t)i / (uint32_t)DHW;
        uint32_t s = (uint32_t)i - b * (uint32_t)DHW;
        const float* dp = disp + (size_t)b * (size_t)(3 * DHW) + s;
        uint32_t l0 = ldsBase + (uint32_t)(((stg * 3 + 0) * TILE + tid) * 4);
        uint32_t l1 = l0 + TILE * 4;
        uint32_t l2 = l1 + TILE * 4;
        uint64_t a0 = (uint64_t)(uintptr_t)(dp);
        uint64_t a1 = (uint64_t)(uintptr_t)(dp + DHW);
        uint64_t a2 = (uint64_t)(uintptr_t)(dp + 2 * DHW);
        asm volatile(
            "global_load_async_to_lds_b32 %0, %1, off\n\t"
            "global_load_async_to_lds_b32 %2, %3, off\n\t"
            "global_load_async_to_lds_b32 %4, %5, off"
            :
            : "v"(l0), "v"(a0), "v"(l1), "v"(a1), "v"(l2), "v"(a2)
            : "memory");
    };

    const int tile0 = (int)blockIdx.x;
    if (tile0 >= numTiles) return;

    // Prologue: prefetch first tile's displacement into stage 0.
    issue_tile(tile0, 0);

    int stg = 0;
    for (int tile = tile0; tile < numTiles; tile += (int)gridDim.x) {
        const int  nextTile = tile + (int)gridDim.x;
        const bool hasNext  = nextTile < numTiles;

        if (hasNext) {
            issue_tile(nextTile, stg ^ 1);
            // 3 ops (next tile) may remain in flight; current tile's 3 retired.
            asm volatile("s_wait_asynccnt 3" ::: "memory");
        } else {
            asm volatile("s_wait_asynccnt 0" ::: "memory");
        }

        const float dz = sdisp[stg][0][tid];
        const float dy = sdisp[stg][1][tid];
        const float dx = sdisp[stg][2][tid];

        const int i = tile * TILE + tid;
        if (i < NPTS) {
            uint32_t b = (uint32_t)i / (uint32_t)DHW;
            uint32_t s = (uint32_t)i - b * (uint32_t)DHW;
            uint32_t z = s / (uint32_t)HW;
            uint32_t r = s - z * (uint32_t)HW;
            uint32_t y = r / (uint32_t)W_;
            uint32_t x = r - y * (uint32_t)W_;

            // align_corners=True normalize/denormalize collapses to x + dx.
            float xs = fminf(fmaxf((float)x + dx, 0.0f), (float)(W_ - 1));
            float ys = fminf(fmaxf((float)y + dy, 0.0f), (float)(H_ - 1));
            float zs = fminf(fmaxf((float)z + dz, 0.0f), (float)(D_ - 1));

            float fx = floorf(xs), fy = floorf(ys), fz = floorf(zs);
            float wx = xs - fx,   wy = ys - fy,   wz = zs - fz;
            int x0 = (int)fx, y0 = (int)fy, z0 = (int)fz;
            int x1 = min(x0 + 1, W_ - 1);
            int y1 = min(y0 + 1, H_ - 1);
            int z1 = min(z0 + 1, D_ - 1);

            uint32_t zo0 = (uint32_t)z0 * HW, zo1 = (uint32_t)z1 * HW;
            uint32_t yo0 = (uint32_t)y0 * W_, yo1 = (uint32_t)y1 * W_;
            uint32_t i000 = zo0 + yo0 + (uint32_t)x0, i001 = zo0 + yo0 + (uint32_t)x1;
            uint32_t i010 = zo0 + yo1 + (uint32_t)x0, i011 = zo0 + yo1 + (uint32_t)x1;
            uint32_t i100 = zo1 + yo0 + (uint32_t)x0, i101 = zo1 + yo0 + (uint32_t)x1;
            uint32_t i110 = zo1 + yo1 + (uint32_t)x0, i111 = zo1 + yo1 + (uint32_t)x1;

            const float* pb = src + (size_t)b * (size_t)(C_ * DHW);
            float*       ob = out + (size_t)b * (size_t)(C_ * DHW);

#pragma unroll
            for (int c = 0; c < C_; ++c) {
                const float* p = pb + (size_t)c * (size_t)DHW;
                // 8 gathers: spatially local (|disp|~2 voxels) -> L1/L2 hits;
                // whole 94MB src lives in the 192MB L2 after first touch.
                float v000 = p[i000], v001 = p[i001];
                float v010 = p[i010], v011 = p[i011];
                float v100 = p[i100], v101 = p[i101];
                float v110 = p[i110], v111 = p[i111];

                float c00 = fmaf(wx, v001 - v000, v000);
                float c01 = fmaf(wx, v011 - v010, v010);
                float c10 = fmaf(wx, v101 - v100, v100);
                float c11 = fmaf(wx, v111 - v110, v110);
                float c0  = fmaf(wy, c01 - c00, c00);
                float c1  = fmaf(wy, c11 - c10, c10);
                float res = fmaf(wz, c1 - c0, c0);

                // Output is write-once: non-temporal so it doesn't evict src from L2.
                __builtin_nontemporal_store(res, ob + (size_t)c * (size_t)DHW + s);
            }
        }
        stg ^= 1;
    }
}

extern "C" void kernel_launch(void* const* d_in, const int* in_sizes, int n_in,
                              void* d_out, int out_size, void* d_ws, size_t ws_size,
                              hipStream_t stream) {
    const float* src  = (const float*)d_in[0];   // [B,C,D,H,W] f32
    const float* disp = (const float*)d_in[1];   // [B,3,D,H,W] f32
    float*       out  = (float*)d_out;           // [B,C,D,H,W] f32

    const int numTiles = (NPTS + TILE - 1) / TILE;      // 46080
    int grid = numTiles < 6144 ? numTiles : 6144;        // persistent blocks, ~7.5 tiles each
    hipLaunchKernelGGL(warp3d_async_kernel, dim3(grid), dim3(TILE), 0, stream,
                       src, disp, out, numTiles);
}